// FLAKimiLinearAttention_24129126269511
// MI455X (gfx1250) — compile-verified
//
#include <hip/hip_runtime.h>
#include <hip/hip_bf16.h>
#include <math.h>

typedef __attribute__((ext_vector_type(2))) float v2f;
typedef __attribute__((ext_vector_type(8))) float v8f;

#define SEQL 1024
#define HID  1024
#define NH   16
#define HD   64
#define CH   64          // chunk length
#define NC   (SEQL/CH)   // 16 chunks
#define TP   65          // LDS pitch (odd -> conflict-free column reads)
#define EPSV 1e-6f

// D = A(16x4 f32) * B(4x16 f32) + C(16x16 f32), full fp32, RNE.
static __device__ __forceinline__ v8f wmma_f32(v2f a, v2f b, v8f c) {
  return __builtin_amdgcn_wmma_f32_16x16x4_f32(false, a, false, b, (short)0, c,
                                               false, false);
}

// ---------------------------------------------------------------------------
// Y[m][n] = sum_k X[m][k] * W[n][k]   (i.e. X @ W.T), optional elu(x)+1.
// Workgroup: 64x64 output tile, 4 waves; wave w owns rows [16w,16w+16) x 64.
// ---------------------------------------------------------------------------
__global__ __launch_bounds__(128) void gemm_xwt(const float* __restrict__ X,
                                                const float* __restrict__ W,
                                                float* __restrict__ Y,
                                                int feature_map) {
  __shared__ float xs[64 * TP];
  __shared__ float wsh[64 * TP];
  const int tid  = threadIdx.x;
  const int wave = tid >> 5;
  const int lane = tid & 31;
  const int n    = lane & 15;
  const int hb   = lane >> 4;
  const int mBase = blockIdx.y * 64;
  const int nBase = blockIdx.x * 64;

  v8f acc[4] = {};

  for (int kc = 0; kc < HID; kc += 64) {
    __syncthreads();
    // Stage X[mBase:+64][kc:+64] and W[nBase:+64][kc:+64] into LDS.
    for (int i = 0; i < 8; ++i) {
      int idx4 = tid + i * 128;          // 1024 float4s per tile
      int row  = idx4 >> 4;
      int c4   = (idx4 & 15) << 2;
      float4 xv = *reinterpret_cast<const float4*>(&X[(mBase + row) * HID + kc + c4]);
      float4 wv = *reinterpret_cast<const float4*>(&W[(nBase + row) * HID + kc + c4]);
      float* xd = &xs[row * TP + c4];
      xd[0] = xv.x; xd[1] = xv.y; xd[2] = xv.z; xd[3] = xv.w;
      float* wd = &wsh[row * TP + c4];
      wd[0] = wv.x; wd[1] = wv.y; wd[2] = wv.z; wd[3] = wv.w;
    }
    __syncthreads();

    const int arow = (wave * 16 + n) * TP;
    for (int kk = 0; kk < 64; kk += 4) {
      v2f a;
      a[0] = xs[arow + kk + 2 * hb];
      a[1] = xs[arow + kk + 2 * hb + 1];
#pragma unroll
      for (int t = 0; t < 4; ++t) {
        v2f b;
        const int brow = (t * 16 + n) * TP;
        b[0] = wsh[brow + kk + 2 * hb];
        b[1] = wsh[brow + kk + 2 * hb + 1];
        acc[t] = wmma_f32(a, b, acc[t]);
      }
    }
  }

#pragma unroll
  for (int t = 0; t < 4; ++t)
#pragma unroll
    for (int r = 0; r < 8; ++r) {
      int row = mBase + wave * 16 + r + 8 * hb;
      int col = nBase + t * 16 + n;
      float yv = acc[t][r];
      if (feature_map) yv = (yv > 0.0f) ? (yv + 1.0f) : __expf(yv);  // elu(x)+1
      Y[row * HID + col] = yv;
    }
}

// ---------------------------------------------------------------------------
// Pass 1: per (head, chunk) compute S_c = K_c^T V_c (64x64) and k column sums.
// ---------------------------------------------------------------------------
__global__ __launch_bounds__(128) void attn_pass1(const float* __restrict__ K,
                                                  const float* __restrict__ V,
                                                  float* __restrict__ Sall,
                                                  float* __restrict__ Kall) {
  __shared__ float ks[64 * TP];
  __shared__ float vs[64 * TP];
  const int c = blockIdx.x, h = blockIdx.y;
  const int tid = threadIdx.x, wave = tid >> 5, lane = tid & 31;
  const int n = lane & 15, hb = lane >> 4;
  const int lBase = c * 64, cb = h * 64;

  for (int i = 0; i < 8; ++i) {
    int idx4 = tid + i * 128;
    int row  = idx4 >> 4;
    int c4   = (idx4 & 15) << 2;
    float4 kv = *reinterpret_cast<const float4*>(&K[(lBase + row) * HID + cb + c4]);
    float4 vv = *reinterpret_cast<const float4*>(&V[(lBase + row) * HID + cb + c4]);
    float* kd = &ks[row * TP + c4];
    kd[0] = kv.x; kd[1] = kv.y; kd[2] = kv.z; kd[3] = kv.w;
    float* vd = &vs[row * TP + c4];
    vd[0] = vv.x; vd[1] = vv.y; vd[2] = vv.z; vd[3] = vv.w;
  }
  __syncthreads();

  // S[d][e] = sum_j K[j][d] * V[j][e];  A = K^T (A[d][j] = K[j][d]).
  v8f acc[4] = {};
  for (int kk = 0; kk < 64; kk += 4) {
    v2f a;
    a[0] = ks[(kk + 2 * hb) * TP + wave * 16 + n];
    a[1] = ks[(kk + 1 + 2 * hb) * TP + wave * 16 + n];
#pragma unroll
    for (int t = 0; t < 4; ++t) {
      v2f b;
      b[0] = vs[(kk + 2 * hb) * TP + t * 16 + n];
      b[1] = vs[(kk + 1 + 2 * hb) * TP + t * 16 + n];
      acc[t] = wmma_f32(a, b, acc[t]);
    }
  }
  float* Sc = &Sall[(size_t)(h * NC + c) << 12];   // 4096 floats per chunk state
#pragma unroll
  for (int t = 0; t < 4; ++t)
#pragma unroll
    for (int r = 0; r < 8; ++r) {
      int d = wave * 16 + r + 8 * hb, e = t * 16 + n;
      Sc[d * 64 + e] = acc[t][r];
    }

  if (tid < 64) {
    float s = 0.0f;
    for (int j = 0; j < 64; ++j) s += ks[j * TP + tid];
    Kall[(h * NC + c) * 64 + tid] = s;
  }
}

// ---------------------------------------------------------------------------
// Pass 2: per (head, chunk): S_prev = sum_{c'<c} S_c', k_prev likewise;
// A = mask(Q K^T); den = rowsum(A) + Q.k_prev; num = Q@S_prev + A@V; out=num/den.
// ---------------------------------------------------------------------------
__global__ __launch_bounds__(128) void attn_pass2(const float* __restrict__ Q,
                                                  const float* __restrict__ K,
                                                  const float* __restrict__ V,
                                                  const float* __restrict__ Sall,
                                                  const float* __restrict__ Kall,
                                                  float* __restrict__ O) {
  extern __shared__ float sm[];
  float* qs  = sm;               // 64*TP
  float* ks  = qs + 64 * TP;
  float* vs  = ks + 64 * TP;
  float* as_ = vs + 64 * TP;
  float* ss  = as_ + 64 * TP;
  float* den = ss + 64 * TP;     // 64
  float* kp  = den + 64;         // 64

  const int c = blockIdx.x, h = blockIdx.y;
  const int tid = threadIdx.x, wave = tid >> 5, lane = tid & 31;
  const int n = lane & 15, hb = lane >> 4;
  const int lBase = c * 64, cb = h * 64;

  for (int i = 0; i < 8; ++i) {
    int idx4 = tid + i * 128;
    int row  = idx4 >> 4;
    int c4   = (idx4 & 15) << 2;
    float4 qv = *reinterpret_cast<const float4*>(&Q[(lBase + row) * HID + cb + c4]);
    float4 kv = *reinterpret_cast<const float4*>(&K[(lBase + row) * HID + cb + c4]);
    float4 vv = *reinterpret_cast<const float4*>(&V[(lBase + row) * HID + cb + c4]);
    float* p;
    p = &qs[row * TP + c4]; p[0] = qv.x; p[1] = qv.y; p[2] = qv.z; p[3] = qv.w;
    p = &ks[row * TP + c4]; p[0] = kv.x; p[1] = kv.y; p[2] = kv.z; p[3] = kv.w;
    p = &vs[row * TP + c4]; p[0] = vv.x; p[1] = vv.y; p[2] = vv.z; p[3] = vv.w;
  }

  // Exclusive prefix of per-chunk states, accumulated in registers.
  {
    float4 sacc[8];
#pragma unroll
    for (int i = 0; i < 8; ++i) sacc[i] = make_float4(0.f, 0.f, 0.f, 0.f);
    const float* Sb = &Sall[(size_t)(h * NC) << 12];
    for (int cp = 0; cp < c; ++cp) {
      const float4* Sp = reinterpret_cast<const float4*>(&Sb[(size_t)cp << 12]);
#pragma unroll
      for (int i = 0; i < 8; ++i) {
        float4 t = Sp[tid + i * 128];
        sacc[i].x += t.x; sacc[i].y += t.y; sacc[i].z += t.z; sacc[i].w += t.w;
      }
    }
#pragma unroll
    for (int i = 0; i < 8; ++i) {
      int e4 = (tid + i * 128) * 4;
      int row = e4 >> 6, col = e4 & 63;
      float* p = &ss[row * TP + col];
      p[0] = sacc[i].x; p[1] = sacc[i].y; p[2] = sacc[i].z; p[3] = sacc[i].w;
    }
  }
  if (tid < 64) {
    float s = 0.0f;
    for (int cp = 0; cp < c; ++cp) s += Kall[(h * NC + cp) * 64 + tid];
    kp[tid] = s;
  }
  __syncthreads();

  // A = mask(Q K^T), inclusive diagonal.
  {
    v8f acc[4] = {};
    const int arow = (wave * 16 + n) * TP;
    for (int kk = 0; kk < 64; kk += 4) {
      v2f a;
      a[0] = qs[arow + kk + 2 * hb];
      a[1] = qs[arow + kk + 2 * hb + 1];
#pragma unroll
      for (int t = 0; t < 4; ++t) {
        v2f b;
        const int brow = (t * 16 + n) * TP;   // B[d][j] = K[j][d]
        b[0] = ks[brow + kk + 2 * hb];
        b[1] = ks[brow + kk + 2 * hb + 1];
        acc[t] = wmma_f32(a, b, acc[t]);
      }
    }
#pragma unroll
    for (int t = 0; t < 4; ++t)
#pragma unroll
      for (int r = 0; r < 8; ++r) {
        int i = wave * 16 + r + 8 * hb, j = t * 16 + n;
        as_[i * TP + j] = (j <= i) ? acc[t][r] : 0.0f;
      }
  }
  __syncthreads();

  if (tid < 64) {
    float s = 0.0f;
    for (int j = 0; j < 64; ++j) s += as_[tid * TP + j];
    for (int d = 0; d < 64; ++d) s += qs[tid * TP + d] * kp[d];
    den[tid] = fmaxf(s, EPSV);
  }
  __syncthreads();

  // num = Q @ S_prev + A @ V ; out = num / den
  {
    v8f acc[4] = {};
    const int arow = (wave * 16 + n) * TP;
    for (int kk = 0; kk < 64; kk += 4) {
      v2f a;
      a[0] = qs[arow + kk + 2 * hb];
      a[1] = qs[arow + kk + 2 * hb + 1];
#pragma unroll
      for (int t = 0; t < 4; ++t) {
        v2f b;
        b[0] = ss[(kk + 2 * hb) * TP + t * 16 + n];
        b[1] = ss[(kk + 1 + 2 * hb) * TP + t * 16 + n];
        acc[t] = wmma_f32(a, b, acc[t]);
      }
    }
    for (int kk = 0; kk < 64; kk += 4) {
      v2f a;
      a[0] = as_[arow + kk + 2 * hb];
      a[1] = as_[arow + kk + 2 * hb + 1];
#pragma unroll
      for (int t = 0; t < 4; ++t) {
        v2f b;
        b[0] = vs[(kk + 2 * hb) * TP + t * 16 + n];
        b[1] = vs[(kk + 1 + 2 * hb) * TP + t * 16 + n];
        acc[t] = wmma_f32(a, b, acc[t]);
      }
    }
#pragma unroll
    for (int t = 0; t < 4; ++t)
#pragma unroll
      for (int r = 0; r < 8; ++r) {
        int i = wave * 16 + r + 8 * hb, e = t * 16 + n;
        O[(lBase + i) * HID + cb + e] = acc[t][r] / den[i];
      }
  }
}

// ---------------------------------------------------------------------------
extern "C" void kernel_launch(void* const* d_in, const int* in_sizes, int n_in,
                              void* d_out, int out_size, void* d_ws, size_t ws_size,
                              hipStream_t stream) {
  (void)in_sizes; (void)n_in; (void)out_size; (void)ws_size;
  const float* x  = (const float*)d_in[0];
  const float* Wq = (const float*)d_in[1];
  const float* Wk = (const float*)d_in[2];
  const float* Wv = (const float*)d_in[3];
  const float* Wo = (const float*)d_in[4];
  float* out = (float*)d_out;

  char* ws = (char*)d_ws;
  float* qb   = (float*)(ws);                          // 4 MB
  float* kb   = (float*)(ws + ((size_t)4 << 20));      // 4 MB
  float* vb   = (float*)(ws + ((size_t)8 << 20));      // 4 MB
  float* ab   = (float*)(ws + ((size_t)12 << 20));     // 4 MB
  float* Sall = (float*)(ws + ((size_t)16 << 20));     // 4 MB (16h*16c*64*64)
  float* Kall = (float*)(ws + ((size_t)20 << 20));     // 64 KB

  dim3 g(16, 16), b(128);
  gemm_xwt<<<g, b, 0, stream>>>(x, Wq, qb, 1);
  gemm_xwt<<<g, b, 0, stream>>>(x, Wk, kb, 1);
  gemm_xwt<<<g, b, 0, stream>>>(x, Wv, vb, 0);

  attn_pass1<<<dim3(NC, NH), b, 0, stream>>>(kb, vb, Sall, Kall);

  const size_t smem = (size_t)(5 * 64 * TP + 128) * sizeof(float);  // ~84 KB
  hipFuncSetAttribute((const void*)attn_pass2,
                      hipFuncAttributeMaxDynamicSharedMemorySize, (int)smem);
  attn_pass2<<<dim3(NC, NH), b, smem, stream>>>(qb, kb, vb, Sall, Kall, ab);

  gemm_xwt<<<g, b, 0, stream>>>(ab, Wo, out, 0);
}